// LinearConstraintWalker_89902255440625
// MI455X (gfx1250) — compile-verified
//
#include <hip/hip_runtime.h>
#include <math.h>

#define BATCH    8192
#define DIM      64
#define NSTEPS   50
#define MCON     256
#define XSTRIDE  ((DIM + 1) * NSTEPS)   // 3250
#define WPB      2                      // waves per block
#define TILEROWS 16

// workspace layout (floats)
#define WS_STA 0                         // 16384 floats: A fragments
#define WS_STB 16384                     //  4096 floats: B0 fragments
#define WS_BC  (16384 + 4096)            //   512 floats: interleaved (b, 1/row_norm)

typedef __attribute__((ext_vector_type(2))) float v2f;
typedef __attribute__((ext_vector_type(8))) float v8f;

__device__ __forceinline__ v8f wmma4(v2f a, v2f b, v8f c) {
  return __builtin_amdgcn_wmma_f32_16x16x4_f32(false, a, false, b, (short)0, c,
                                               false, false);
}
__device__ __forceinline__ v2f mk2(float a, float b) { v2f r; r.x = a; r.y = b; return r; }

// ---------------------------------------------------------------------------
// Prep: 1/||A@B0||_rows and WMMA B-operand fragment swizzles of A and B0.
// Fragment element for (ntile n, ktile t, lane, j):
//   value = Mat[16*n + (lane&15)][4*t + 2*(lane>>4) + j]
// packed per k-tile-pair p=t>>1 as float4/lane: col = 8p + 2hi + (j4&1) + 4*(j4>>1)
// ---------------------------------------------------------------------------
__global__ void lcw_prep_kernel(const float* __restrict__ A,
                                const float* __restrict__ b,
                                const float* __restrict__ B0,
                                float* __restrict__ ws) {
  const int tid = threadIdx.x;

  for (int r = tid; r < MCON; r += blockDim.x) {
    float nrm2 = 0.f;
    for (int j = 0; j < DIM; ++j) {
      float d = 0.f;
      for (int k = 0; k < DIM; ++k) d = fmaf(A[r * DIM + k], B0[k * DIM + j], d);
      nrm2 = fmaf(d, d, nrm2);
    }
    ws[WS_BC + 2 * r]     = b[r];
    ws[WS_BC + 2 * r + 1] = 1.0f / sqrtf(nrm2);
  }
  for (int idx = tid; idx < 16384; idx += blockDim.x) {   // A: 16 ntiles
    int j4 = idx & 3, lane = (idx >> 2) & 31, np = idx >> 7;
    int n = np >> 3, p = np & 7, lo = lane & 15, hi = lane >> 4;
    ws[WS_STA + idx] = A[(16 * n + lo) * DIM + 8 * p + 2 * hi + (j4 & 1) + 4 * (j4 >> 1)];
  }
  for (int idx = tid; idx < 4096; idx += blockDim.x) {    // B0: 4 ntiles
    int j4 = idx & 3, lane = (idx >> 2) & 31, np = idx >> 7;
    int n = np >> 3, p = np & 7, lo = lane & 15, hi = lane >> 4;
    ws[WS_STB + idx] = B0[(16 * n + lo) * DIM + 8 * p + 2 * hi + (j4 & 1) + 4 * (j4 >> 1)];
  }
}

// ---------------------------------------------------------------------------
// Walker: one wave32 per 16 batch rows; fp32 WMMA 16x16x4, dual-tile chains.
// ---------------------------------------------------------------------------
__launch_bounds__(WPB * 32, 1)
__global__ void lcw_walker_kernel(const float* __restrict__ x,
                                  const float* __restrict__ x00,
                                  const float* __restrict__ ws,
                                  float* __restrict__ out) {
  __shared__ float sA[16384];          // 64 KB : A fragments
  __shared__ float sB[4096];           // 16 KB : B0 fragments
  __shared__ float sBC[512];           //  2 KB : (b, 1/rn) pairs per column
  __shared__ float x0t[WPB][16 * 68];  // per-wave x0 tile (stride 68: no bank conflicts)
  __shared__ float sbL[WPB][16];       // per-wave scale*beta/||v|| per row
  __shared__ float scl[WPB][16];       // per-wave new scale per row

  const int tid  = threadIdx.x;
  const int w    = tid >> 5;
  const int lane = tid & 31;
  const int lo   = lane & 15;
  const int hi   = lane >> 4;
  const int R    = (blockIdx.x * WPB + w) * TILEROWS;

  for (int i = tid; i < 16384; i += blockDim.x) sA[i] = ws[WS_STA + i];
  for (int i = tid; i < 4096;  i += blockDim.x) sB[i] = ws[WS_STB + i];
  for (int i = tid; i < 512;   i += blockDim.x) sBC[i] = ws[WS_BC + i];
  __syncthreads();

  // x0 state in WMMA C/D layout: lane holds col 16n+lo, rows (8*hi + i)
  float x0d[4][8];
#pragma unroll
  for (int n = 0; n < 4; ++n) {
    float xv = x00[16 * n + lo];
#pragma unroll
    for (int i = 0; i < 8; ++i) x0d[n][i] = xv;
  }
  float scale = 1.0f;

  const float* zrow = x + (size_t)(R + lo) * XSTRIDE;

  for (int s = 0; s < NSTEPS; ++s) {
    // ---- raw v in A-fragment layout + row norm (1/||v|| folded into sb) ----
    const float* zp = zrow + s * (DIM + 1) + 2 * hi;
    v2f uf[16];
    float n2 = 0.f;
#pragma unroll
    for (int t = 0; t < 16; ++t) {
      float a0 = zp[4 * t], a1 = zp[4 * t + 1];
      uf[t] = mk2(a0, a1);
      n2 = fmaf(a0, a0, fmaf(a1, a1, n2));
    }
    n2 += __shfl_xor(n2, 16, 32);
    float inv = 1.0f / fmaxf(sqrtf(n2), 1e-12f);

    // both lanes of a pair compute identical value; duplicate store is benign
    float braw = zrow[s * (DIM + 1) + DIM];
    float beta = 1.0f / (1.0f + __expf(-braw));
    sbL[w][lo] = scale * beta * inv;
    __builtin_amdgcn_wave_barrier();
    float sb[8];
#pragma unroll
    for (int i = 0; i < 8; ++i) sb[i] = sbL[w][8 * hi + i];

    // ---- x0 += sb * (v @ B0^T): 2 pairs of ntiles, dual WMMA chains ----
#pragma unroll
    for (int n = 0; n < 4; n += 2) {
      v8f acc0 = {}, acc1 = {};
#pragma unroll
      for (int p = 0; p < 8; ++p) {
        const float4 f0 = *(const float4*)&sB[(((n * 8 + p) * 32) + lane) * 4];
        const float4 f1 = *(const float4*)&sB[((((n + 1) * 8 + p) * 32) + lane) * 4];
        acc0 = wmma4(uf[2 * p],     mk2(f0.x, f0.y), acc0);
        acc1 = wmma4(uf[2 * p],     mk2(f1.x, f1.y), acc1);
        acc0 = wmma4(uf[2 * p + 1], mk2(f0.z, f0.w), acc0);
        acc1 = wmma4(uf[2 * p + 1], mk2(f1.z, f1.w), acc1);
      }
#pragma unroll
      for (int i = 0; i < 8; ++i) {
        x0d[n][i]     = fmaf(sb[i], acc0[i], x0d[n][i]);
        x0d[n + 1][i] = fmaf(sb[i], acc1[i], x0d[n + 1][i]);
        x0t[w][(8 * hi + i) * 68 + 16 * n + lo]       = x0d[n][i];
        x0t[w][(8 * hi + i) * 68 + 16 * (n + 1) + lo] = x0d[n + 1][i];
      }
    }
    __builtin_amdgcn_wave_barrier();

    // ---- reload x0 as A-fragments (transpose via LDS) ----
    v2f xf[16];
#pragma unroll
    for (int t = 0; t < 16; ++t) {
      const float* xp = &x0t[w][lo * 68 + 4 * t + 2 * hi];
      xf[t] = mk2(xp[0], xp[1]);
    }

    // ---- scale = min_m (b - x0@A^T)/rn : 8 pairs of ntiles, dual chains ----
    float rowmin[8];
#pragma unroll
    for (int i = 0; i < 8; ++i) rowmin[i] = INFINITY;
#pragma unroll
    for (int n = 0; n < 16; n += 2) {
      v8f acc0 = {}, acc1 = {};
#pragma unroll
      for (int p = 0; p < 8; ++p) {
        const float4 f0 = *(const float4*)&sA[(((n * 8 + p) * 32) + lane) * 4];
        const float4 f1 = *(const float4*)&sA[((((n + 1) * 8 + p) * 32) + lane) * 4];
        acc0 = wmma4(xf[2 * p],     mk2(f0.x, f0.y), acc0);
        acc1 = wmma4(xf[2 * p],     mk2(f1.x, f1.y), acc1);
        acc0 = wmma4(xf[2 * p + 1], mk2(f0.z, f0.w), acc0);
        acc1 = wmma4(xf[2 * p + 1], mk2(f1.z, f1.w), acc1);
      }
      const float2 c0 = *(const float2*)&sBC[2 * (16 * n + lo)];
      const float2 c1 = *(const float2*)&sBC[2 * (16 * (n + 1) + lo)];
#pragma unroll
      for (int i = 0; i < 8; ++i) {
        rowmin[i] = fminf(rowmin[i], (c0.x - acc0[i]) * c0.y);
        rowmin[i] = fminf(rowmin[i], (c1.x - acc1[i]) * c1.y);
      }
    }
    // min across the 16 lanes sharing each row half
#pragma unroll
    for (int d = 1; d <= 8; d <<= 1)
#pragma unroll
      for (int i = 0; i < 8; ++i)
        rowmin[i] = fminf(rowmin[i], __shfl_xor(rowmin[i], d, 32));
    if (lo == 0) {
#pragma unroll
      for (int i = 0; i < 8; ++i) scl[w][8 * hi + i] = rowmin[i];
    }
    __builtin_amdgcn_wave_barrier();
    scale = scl[w][lo];
  }

  // ---- write x0 (D-layout) to out[row*64 + col] ----
#pragma unroll
  for (int n = 0; n < 4; ++n)
#pragma unroll
    for (int i = 0; i < 8; ++i)
      out[(size_t)(R + 8 * hi + i) * DIM + 16 * n + lo] = x0d[n][i];
}

extern "C" void kernel_launch(void* const* d_in, const int* in_sizes, int n_in,
                              void* d_out, int out_size, void* d_ws, size_t ws_size,
                              hipStream_t stream) {
  const float* x   = (const float*)d_in[0];
  const float* A   = (const float*)d_in[1];
  const float* b   = (const float*)d_in[2];
  const float* B0  = (const float*)d_in[3];
  const float* x00 = (const float*)d_in[4];
  float* out = (float*)d_out;
  float* ws  = (float*)d_ws;

  lcw_prep_kernel<<<dim3(1), dim3(256), 0, stream>>>(A, b, B0, ws);
  lcw_walker_kernel<<<dim3(BATCH / (WPB * TILEROWS)), dim3(WPB * 32), 0, stream>>>(
      x, x00, ws, out);
}